// Model_51453708206388
// MI455X (gfx1250) — compile-verified
//
#include <hip/hip_runtime.h>

// Problem dims (B, H, S_MAX, S_NEW, D) = (4, 8, 8192, 2048, 128)
#define B_    4
#define H_    8
#define SMAX  8192
#define SNEW  2048
#define D_    128
#define HALF  64

// Output layout (flat bf16 elements, reference return order)
#define N_KS   (B_*H_*SNEW*D_)      // 8,388,608  grad_key_states
#define N_VS   (B_*H_*SNEW*D_)      // 8,388,608  grad_value_states
#define N_CS   (B_*SNEW*D_)         // 1,048,576  grad_cos
#define N_SN   (B_*SNEW*D_)         // 1,048,576  grad_sin
#define N_KC   (B_*H_*SMAX*D_)      // 33,554,432 grad_key_cache_input
#define OFF1   0
#define OFF2   (OFF1 + N_KS)
#define OFF3   (OFF2 + N_VS)
#define OFF4   (OFF3 + N_CS)
#define OFF5   (OFF4 + N_SN)
#define OFF6   (OFF5 + N_KC)

typedef __attribute__((ext_vector_type(2)))  float        v2f;
typedef __attribute__((ext_vector_type(8)))  float        v8f;
typedef __attribute__((ext_vector_type(4)))  float        f4;
typedef __attribute__((ext_vector_type(2)))  unsigned int u2;
typedef __attribute__((ext_vector_type(4)))  unsigned int u4;

// ---------------------------------------------------------------------------
// float pair -> packed bf16x2, round-to-nearest-even. Bit-exact with
// jax astype(bfloat16) for all finite values (inputs here are gaussian
// normals, so the whole pipeline is finite; Inf would survive the RNE add
// untouched anyway). Pack of the two rounded high halves is one v_perm_b32:
// sel 0x07060302 -> { hi_r[3], hi_r[2], lo_r[3], lo_r[2] }.
// 5 VALU ops per pair (2x v_bfe + 2x v_add3 + 1x v_perm).
// ---------------------------------------------------------------------------
#if __has_builtin(__builtin_amdgcn_cvt_pk_bf16_f32)
__device__ __forceinline__ unsigned pack_bf16(float lo, float hi) {
    typedef __attribute__((ext_vector_type(2))) __bf16 v2bf;
    v2bf v = __builtin_amdgcn_cvt_pk_bf16_f32(lo, hi);
    return __builtin_bit_cast(unsigned, v);
}
#else
__device__ __forceinline__ unsigned rne_bf16_hi(float f) {
    unsigned u = __float_as_uint(f);
    return u + 0x7fffu + ((u >> 16) & 1u);     // rounded bf16 in bits [31:16]
}
__device__ __forceinline__ unsigned pack_bf16(float lo, float hi) {
    return __builtin_amdgcn_perm(rne_bf16_hi(hi), rne_bf16_hi(lo), 0x07060302u);
}
#endif

// ---------------------------------------------------------------------------
// K1: stream-convert full caches f32 -> bf16 (outputs 5 & 6), non-temporal.
// 268 MB read + 134 MB write — the dominant roofline term (~17 us @ 23.3 TB/s).
// 8 elements per array per thread: 2x global_load_b128 + 1x store_b128 each.
// ---------------------------------------------------------------------------
__global__ __launch_bounds__(256) void k_stream_cvt(
    const f4* __restrict__ G4, const f4* __restrict__ V4,
    u4* __restrict__ out5, u4* __restrict__ out6)
{
    unsigned i = blockIdx.x * 256u + threadIdx.x;        // exact grid, no tail
    f4 a0 = __builtin_nontemporal_load(G4 + 2 * i);
    f4 a1 = __builtin_nontemporal_load(G4 + 2 * i + 1);
    f4 c0 = __builtin_nontemporal_load(V4 + 2 * i);
    f4 c1 = __builtin_nontemporal_load(V4 + 2 * i + 1);
    u4 pa, pc;
    pa.x = pack_bf16(a0.x, a0.y); pa.y = pack_bf16(a0.z, a0.w);
    pa.z = pack_bf16(a1.x, a1.y); pa.w = pack_bf16(a1.z, a1.w);
    pc.x = pack_bf16(c0.x, c0.y); pc.y = pack_bf16(c0.z, c0.w);
    pc.z = pack_bf16(c1.x, c1.y); pc.w = pack_bf16(c1.z, c1.w);
    __builtin_nontemporal_store(pa, out5 + i);
    __builtin_nontemporal_store(pc, out6 + i);
}

// ---------------------------------------------------------------------------
// K2: gathered rows. thread <-> (b,h,s,u), u in [0,16): owns d=4u..4u+3 and
// d+64, so the rotate-half pairing is fully in-register. f4 loads, packed u2
// stores. Also zeroes the gathered rows of outputs 5/6 (after K1, same stream).
// ---------------------------------------------------------------------------
__global__ __launch_bounds__(256) void k_gather_elem(
    const float* __restrict__ G, const float* __restrict__ V,
    const float* __restrict__ Cos, const float* __restrict__ Sin,
    const int* __restrict__ cpos,
    unsigned short* __restrict__ out1, unsigned short* __restrict__ out2,
    unsigned short* __restrict__ out5, unsigned short* __restrict__ out6)
{
    unsigned idx = blockIdx.x * 256u + threadIdx.x;      // < 1,048,576
    unsigned u = idx & 15u;
    unsigned s = (idx >> 4) & (SNEW - 1);
    unsigned h = (idx >> 15) & (H_ - 1);
    unsigned b = idx >> 18;
    int pos = cpos[s];

    size_t cbase = (((size_t)(b * H_ + h)) * SMAX + (size_t)pos) * D_ + 4 * u;
    size_t rbase = ((size_t)b * SNEW + s) * D_ + 4 * u;
    size_t obase = (((size_t)(b * H_ + h)) * SNEW + s) * D_ + 4 * u;

    f4 gl = *(const f4*)(G + cbase);        f4 gh = *(const f4*)(G + cbase + HALF);
    f4 vl = *(const f4*)(V + cbase);        f4 vh = *(const f4*)(V + cbase + HALF);
    f4 cl = *(const f4*)(Cos + rbase);      f4 ch = *(const f4*)(Cos + rbase + HALF);
    f4 sl = *(const f4*)(Sin + rbase);      f4 sh = *(const f4*)(Sin + rbase + HALF);

    // grad_key_states: lo = g_lo*cos_lo + g_hi*sin_hi ; hi = g_hi*cos_hi - g_lo*sin_lo
    f4 ol = gl * cl + gh * sh;
    f4 oh = gh * ch - gl * sl;

    u2 zero = {0u, 0u};
    *(u2*)(out1 + obase)        = (u2){ pack_bf16(ol.x, ol.y), pack_bf16(ol.z, ol.w) };
    *(u2*)(out1 + obase + HALF) = (u2){ pack_bf16(oh.x, oh.y), pack_bf16(oh.z, oh.w) };
    *(u2*)(out2 + obase)        = (u2){ pack_bf16(vl.x, vl.y), pack_bf16(vl.z, vl.w) };
    *(u2*)(out2 + obase + HALF) = (u2){ pack_bf16(vh.x, vh.y), pack_bf16(vh.z, vh.w) };
    *(u2*)(out5 + cbase)        = zero;
    *(u2*)(out5 + cbase + HALF) = zero;
    *(u2*)(out6 + cbase)        = zero;
    *(u2*)(out6 + cbase + HALF) = zero;
}

// ---------------------------------------------------------------------------
// K3: grad_cos/grad_sin = sum over H=8 of g*k / g*rot(k), computed with
// V_WMMA_F32_16X16X4_F32 (B = all-ones): D[m,n] = sum_k A[m,k] + C.
// A-matrix lane layout (ISA 7.12.2): lane L -> M = L&15, K = 2*(L>>4)+{0,1}.
// Two chained WMMAs (k->h and k->h+4) cover all 8 heads in f32 precision.
// One wave per (b,s); d processed in 16-wide blocks (rows of the D matrix).
// ---------------------------------------------------------------------------
__global__ __launch_bounds__(128) void k_reduce_wmma(
    const float* __restrict__ G, const float* __restrict__ K,
    const int* __restrict__ cpos,
    unsigned short* __restrict__ out3, unsigned short* __restrict__ out4)
{
    const int lane = threadIdx.x;                 // 0..31 (wave32)
    const int wave = threadIdx.y;                 // 0..3
    const int bs   = blockIdx.x * 4 + wave;       // 0..8191 (exact)
    const int b    = bs >> 11;                    // / SNEW
    const int s    = bs & (SNEW - 1);
    const int pos  = cpos[s];
    const int m    = lane & 15;
    const int q    = lane >> 4;

    const size_t HSG = (size_t)SMAX * D_;         // h-stride in caches
    const size_t HSK = (size_t)SNEW * D_;         // h-stride in key_states
    const size_t gBase = ((size_t)b * H_ * SMAX + (size_t)pos) * D_;
    const size_t kBase = ((size_t)b * H_ * SNEW + (size_t)s) * D_;

    const v2f ones = {1.0f, 1.0f};

    for (int d0 = 0; d0 < D_; d0 += 16) {
        const int d  = d0 + m;
        const int dp = d ^ HALF;                  // rotate-half partner
        const float sign = (d < HALF) ? -1.0f : 1.0f;

        v8f ccos = {0.f,0.f,0.f,0.f,0.f,0.f,0.f,0.f};
        v8f csin = {0.f,0.f,0.f,0.f,0.f,0.f,0.f,0.f};

        #pragma unroll
        for (int j = 0; j < 2; ++j) {
            const int h0 = 2 * q + 4 * j;         // this lane's K-slots -> heads
            float gx = G[gBase + (size_t)h0       * HSG + d];
            float gy = G[gBase + (size_t)(h0 + 1) * HSG + d];
            float kx = K[kBase + (size_t)h0       * HSK + d];
            float ky = K[kBase + (size_t)(h0 + 1) * HSK + d];
            float px = K[kBase + (size_t)h0       * HSK + dp];
            float py = K[kBase + (size_t)(h0 + 1) * HSK + dp];
            v2f a_cos = { gx * kx,        gy * ky        };
            v2f a_sin = { sign * gx * px, sign * gy * py };
            ccos = __builtin_amdgcn_wmma_f32_16x16x4_f32(
                       false, a_cos, false, ones, (short)0, ccos, false, false);
            csin = __builtin_amdgcn_wmma_f32_16x16x4_f32(
                       false, a_sin, false, ones, (short)0, csin, false, false);
        }

        // D layout: lane 0 VGPR j -> (M=j, N=0); lane 16 VGPR j -> (M=8+j, N=0).
        // All columns identical (B = ones), so lanes 0 and 16 hold the 16 sums.
        if (m == 0) {
            const size_t r = ((size_t)b * SNEW + s) * D_ + d0 + q * 8;
            u4 pc, ps;
            pc.x = pack_bf16(ccos[0], ccos[1]); pc.y = pack_bf16(ccos[2], ccos[3]);
            pc.z = pack_bf16(ccos[4], ccos[5]); pc.w = pack_bf16(ccos[6], ccos[7]);
            ps.x = pack_bf16(csin[0], csin[1]); ps.y = pack_bf16(csin[2], csin[3]);
            ps.z = pack_bf16(csin[4], csin[5]); ps.w = pack_bf16(csin[6], csin[7]);
            *(u4*)(out3 + r) = pc;
            *(u4*)(out4 + r) = ps;
        }
    }
}

// ---------------------------------------------------------------------------
extern "C" void kernel_launch(void* const* d_in, const int* in_sizes, int n_in,
                              void* d_out, int out_size, void* d_ws, size_t ws_size,
                              hipStream_t stream) {
    (void)in_sizes; (void)n_in; (void)out_size; (void)d_ws; (void)ws_size;

    const float* gkc  = (const float*)d_in[0];   // grad_key_cache   (B,H,SMAX,D) f32
    const float* gvc  = (const float*)d_in[1];   // grad_value_cache (B,H,SMAX,D) f32
    const float* key  = (const float*)d_in[2];   // key_states       (B,H,SNEW,D) f32
    const float* cosp = (const float*)d_in[3];   // cos              (B,SNEW,D)   f32
    const float* sinp = (const float*)d_in[4];   // sin              (B,SNEW,D)   f32
    const int*   cpos = (const int*)d_in[5];     // cache_position   (SNEW,)      i32

    unsigned short* out  = (unsigned short*)d_out;
    unsigned short* out1 = out + OFF1;
    unsigned short* out2 = out + OFF2;
    unsigned short* out3 = out + OFF3;
    unsigned short* out4 = out + OFF4;
    unsigned short* out5 = out + OFF5;
    unsigned short* out6 = out + OFF6;

    // K1: full cache conversion, 8 elems per array per thread: 4,194,304 threads
    k_stream_cvt<<<N_KC / 8 / 256, 256, 0, stream>>>(
        (const f4*)gkc, (const f4*)gvc, (u4*)out5, (u4*)out6);

    // K2: gathered elementwise + zeroing, vec4: B*H*SNEW*16 = 1,048,576 threads
    k_gather_elem<<<(B_*H_*SNEW*16) / 256, 256, 0, stream>>>(
        gkc, gvc, cosp, sinp, cpos, out1, out2, out5, out6);

    // K3: WMMA H-reduction: one wave per (b,s); (32,4) blocks, 2048 blocks
    dim3 blk(32, 4, 1);
    k_reduce_wmma<<<(B_*SNEW) / 4, blk, 0, stream>>>(gkc, key, cpos, out3, out4);
}